// UniGNN_18081812316776
// MI455X (gfx1250) — compile-verified
//
#include <hip/hip_runtime.h>

// ---------------- problem constants (fixed-shape harness) ----------------
#define N_NODES 100000
#define N_EDGES 20000
#define NNZ     800000
#define F_IN    128
#define HID     128
#define F_OUT   64

typedef float v2f __attribute__((ext_vector_type(2)));
typedef float v8f __attribute__((ext_vector_type(8)));

// ---------------- utility kernels ----------------
__global__ void zero_f32(float* __restrict__ p, int n) {
    int t = blockIdx.x * blockDim.x + threadIdx.x;
    if (t < n) p[t] = 0.0f;
}

__global__ void edge_count(const int* __restrict__ edges, float* __restrict__ cnt, int nnz) {
    int t = blockIdx.x * blockDim.x + threadIdx.x;
    if (t < nnz) atomicAdd(&cnt[edges[t]], 1.0f);
}

__global__ void recip_clamp(float* __restrict__ cnt, int e) {
    int t = blockIdx.x * blockDim.x + threadIdx.x;
    if (t < e) cnt[t] = 1.0f / fmaxf(cnt[t], 1.0f);
}

// Xe = esum * rinv[edge]   (in place; rinv holds 1/max(cnt,1))
__global__ void scale_rows(float* __restrict__ buf, const float* __restrict__ rinv, int nrows, int d) {
    int t = blockIdx.x * blockDim.x + threadIdx.x;
    if (t < nrows * d) buf[t] *= rinv[t / d];
}

// out = (1+eps)*Xp + Xv  (+ optional relu); elementwise => in-place safe
__global__ void combine(const float* __restrict__ Xp, const float* __restrict__ Xv,
                        const float* __restrict__ eps, float* __restrict__ out,
                        int n, int do_relu) {
    int t = blockIdx.x * blockDim.x + threadIdx.x;
    if (t >= n) return;
    float s = (1.0f + eps[0]) * Xp[t] + Xv[t];
    if (do_relu) s = fmaxf(s, 0.0f);
    out[t] = s;
}

// dst[sidx[i]] += src[gidx[i]]  rowwise, float4-vectorized, atomics resolve in L2
template<int D>
__global__ void gather_scatter_add(const float* __restrict__ src,
                                   const int* __restrict__ gidx,
                                   const int* __restrict__ sidx,
                                   float* __restrict__ dst, int nnz) {
    constexpr int QPR = D / 4;                 // float4 quads per row
    int t = blockIdx.x * blockDim.x + threadIdx.x;
    int i = t / QPR;
    int q = t % QPR;
    if (i >= nnz) return;
    int g = gidx[i];
    int s = sidx[i];
    const float4 v = *(const float4*)(src + (size_t)g * D + q * 4);
    float* p = dst + (size_t)s * D + q * 4;
    atomicAdd(p + 0, v.x);
    atomicAdd(p + 1, v.y);
    atomicAdd(p + 2, v.z);
    atomicAdd(p + 3, v.w);
}

// ---------------- WMMA fp32 GEMM:  P[N,DOUT] = H[N,DIN] @ W[DIN,DOUT] ----------------
// wave32; one wave per 16-row M-tile; all DOUT/16 N-tiles accumulated in VGPRs.
// A layout (16x4 f32): lanes 0-15 -> K = k0,k0+1 ; lanes 16-31 -> K = k0+2,k0+3 (aligned float2 load)
// B layout (4x16 f32): VGPR0: K=k0 (lanes0-15) / K=k0+2 (lanes16-31); VGPR1: K=k0+1 / k0+3
//   -> W is staged in LDS K-pair-interleaved: ldsW[(p*DOUT + n)*2 + (k&1)], p = k>>1,
//      so each lane's (W[k][n], W[k+1][n]) fragment is ONE aligned ds_load_b64 straight
//      into the even-aligned VGPR pair WMMA SRC1 needs (no v_mov marshaling).
// C layout: VGPR v -> row m0+v (lanes0-15) / m0+v+8 (lanes16-31), col n0+lane%16
template<int DIN, int DOUT>
__global__ __launch_bounds__(256)
void gemm_wmma_f32(const float* __restrict__ H, const float* __restrict__ W,
                   float* __restrict__ P, int nrows) {
    __shared__ float ldsW[DIN * DOUT];
    for (int i = threadIdx.x; i < DIN * DOUT; i += 256) {
        int k = i / DOUT;
        int n = i % DOUT;
        ldsW[((k >> 1) * DOUT + n) * 2 + (k & 1)] = W[i];
    }
    __syncthreads();

    const int wave = threadIdx.x >> 5;
    const int lane = threadIdx.x & 31;
    const int half = lane >> 4;      // 0: lanes 0-15, 1: lanes 16-31
    const int lm   = lane & 15;
    const int mtile = blockIdx.x * 8 + wave;
    if (mtile * 16 >= nrows) return;
    const int m0 = mtile * 16;

    constexpr int NT = DOUT / 16;
    v8f acc[NT];
    v8f vz = {0.f, 0.f, 0.f, 0.f, 0.f, 0.f, 0.f, 0.f};
#pragma unroll
    for (int t = 0; t < NT; ++t) acc[t] = vz;

    const float* arow = H + (size_t)(m0 + lm) * DIN + 2 * half;   // 8B-aligned
    const v2f* bpairs = (const v2f*)ldsW;                         // pair-interleaved view
#pragma unroll
    for (int k0 = 0; k0 < DIN; k0 += 4) {
        v2f a = *(const v2f*)(arow + k0);
        const int pb = ((k0 >> 1) + half) * DOUT + lm;            // K-pair row for this half
#pragma unroll
        for (int t = 0; t < NT; ++t) {
            v2f b = bpairs[pb + t * 16];                          // single ds_load_b64
            acc[t] = __builtin_amdgcn_wmma_f32_16x16x4_f32(
                false, a, false, b, (short)0, acc[t], false, false);
        }
    }

#pragma unroll
    for (int t = 0; t < NT; ++t) {
#pragma unroll
        for (int v = 0; v < 8; ++v) {
            P[(size_t)(m0 + v + 8 * half) * DOUT + t * 16 + lm] = acc[t][v];
        }
    }
}

// ---------------- driver ----------------
static inline int cdiv(int a, int b) { return (a + b - 1) / b; }

extern "C" void kernel_launch(void* const* d_in, const int* in_sizes, int n_in,
                              void* d_out, int out_size, void* d_ws, size_t ws_size,
                              hipStream_t stream) {
    const float* X      = (const float*)d_in[0];
    const int*   vertex = (const int*)d_in[1];
    const int*   edges  = (const int*)d_in[2];
    const float* W1     = (const float*)d_in[3];
    const float* W2     = (const float*)d_in[4];
    const float* W3     = (const float*)d_in[5];
    const float* eps1   = (const float*)d_in[6];
    const float* eps2   = (const float*)d_in[7];
    const float* eps3   = (const float*)d_in[8];

    float* out   = (float*)d_out;                  // [N_NODES, F_OUT]
    float* XeOut = out + (size_t)N_NODES * F_OUT;  // [N_EDGES, F_OUT]

    float* bufA = (float*)d_ws;                       // N*128
    float* bufB = bufA + (size_t)N_NODES * HID;       // N*128
    float* bufE = bufB + (size_t)N_NODES * HID;       // E*128
    float* rinv = bufE + (size_t)N_EDGES * HID;       // E

    const int T = 256;
    const int gemm_blocks = cdiv(N_NODES / 16, 8);    // one wave per 16-row tile, 8 waves/block
    const int sc128 = cdiv(NNZ * (HID / 4), T);
    const int sc64  = cdiv(NNZ * (F_OUT / 4), T);

    // ---- edge membership counts -> reciprocal (same for all layers) ----
    zero_f32<<<cdiv(N_EDGES, T), T, 0, stream>>>(rinv, N_EDGES);
    edge_count<<<cdiv(NNZ, T), T, 0, stream>>>(edges, rinv, NNZ);
    recip_clamp<<<cdiv(N_EDGES, T), T, 0, stream>>>(rinv, N_EDGES);

    // ================= Layer 1: Xp1 = X @ W1 =================
    gemm_wmma_f32<F_IN, HID><<<gemm_blocks, T, 0, stream>>>(X, W1, bufA, N_NODES);
    zero_f32<<<cdiv(N_EDGES * HID, T), T, 0, stream>>>(bufE, N_EDGES * HID);
    gather_scatter_add<HID><<<sc128, T, 0, stream>>>(bufA, vertex, edges, bufE, NNZ);
    scale_rows<<<cdiv(N_EDGES * HID, T), T, 0, stream>>>(bufE, rinv, N_EDGES, HID);
    zero_f32<<<cdiv(N_NODES * HID, T), T, 0, stream>>>(bufB, N_NODES * HID);
    gather_scatter_add<HID><<<sc128, T, 0, stream>>>(bufE, edges, vertex, bufB, NNZ);
    combine<<<cdiv(N_NODES * HID, T), T, 0, stream>>>(bufA, bufB, eps1, bufA, N_NODES * HID, 1); // H1 in bufA

    // ================= Layer 2: Xp2 = H1 @ W2 =================
    gemm_wmma_f32<HID, HID><<<gemm_blocks, T, 0, stream>>>(bufA, W2, bufB, N_NODES);
    zero_f32<<<cdiv(N_EDGES * HID, T), T, 0, stream>>>(bufE, N_EDGES * HID);
    gather_scatter_add<HID><<<sc128, T, 0, stream>>>(bufB, vertex, edges, bufE, NNZ);
    scale_rows<<<cdiv(N_EDGES * HID, T), T, 0, stream>>>(bufE, rinv, N_EDGES, HID);
    zero_f32<<<cdiv(N_NODES * HID, T), T, 0, stream>>>(bufA, N_NODES * HID); // H1 only needed by GEMM above
    gather_scatter_add<HID><<<sc128, T, 0, stream>>>(bufE, edges, vertex, bufA, NNZ);
    combine<<<cdiv(N_NODES * HID, T), T, 0, stream>>>(bufB, bufA, eps2, bufB, N_NODES * HID, 1); // H2 in bufB

    // ================= Layer 3: Xp3 = H2 @ W3 (DOUT=64) =================
    float* Xp3 = bufA;                               // N*64
    float* Xv3 = bufA + (size_t)N_NODES * F_OUT;     // N*64 (second half of bufA)
    gemm_wmma_f32<HID, F_OUT><<<gemm_blocks, T, 0, stream>>>(bufB, W3, Xp3, N_NODES);
    // edge sums accumulate straight into the Xe output region
    zero_f32<<<cdiv(N_EDGES * F_OUT, T), T, 0, stream>>>(XeOut, N_EDGES * F_OUT);
    gather_scatter_add<F_OUT><<<sc64, T, 0, stream>>>(Xp3, vertex, edges, XeOut, NNZ);
    scale_rows<<<cdiv(N_EDGES * F_OUT, T), T, 0, stream>>>(XeOut, rinv, N_EDGES, F_OUT); // final Xe
    zero_f32<<<cdiv(N_NODES * F_OUT, T), T, 0, stream>>>(Xv3, N_NODES * F_OUT);
    gather_scatter_add<F_OUT><<<sc64, T, 0, stream>>>(XeOut, edges, vertex, Xv3, NNZ);
    combine<<<cdiv(N_NODES * F_OUT, T), T, 0, stream>>>(Xp3, Xv3, eps3, out, N_NODES * F_OUT, 0);
}